// GraphEncoder_51436528337076
// MI455X (gfx1250) — compile-verified
//
#include <hip/hip_runtime.h>

// Problem dimensions (fixed by the reference).
#define N_NODES 100000
#define HID     128
#define NE      500000
#define NP      100000
#define E_TOTAL (NE + NP)

#define EB   4    // edges per async batch per wave
#define AWPB 8    // waves per block in aggregate
#define SCAN_T 1024

// ---- helpers -------------------------------------------------------------

// Flat (generic) pointer -> 32-bit LDS byte offset, for async-load VDST.
__device__ __forceinline__ unsigned lds_addr_of(const void* p) {
  return (unsigned)(unsigned long long)(const __attribute__((address_space(3))) void*)p;
}

// row/col of concatenated edge list: [edge_index | prior_index], each [2, E] flat.
__device__ __forceinline__ void edge_rc(const int* __restrict__ eidx,
                                        const int* __restrict__ pidx,
                                        int e, int& r, int& c) {
  if (e < NE) { r = eidx[e];    c = eidx[NE + e]; }
  else        { int t = e - NE; r = pidx[t]; c = pidx[NP + t]; }
}

// ---- pull-style aggregation (hot kernel, placed first for disasm) --------
// One wave per destination node. Incoming feature rows are async-gathered
// into LDS (32 lanes x b128 = one 512B row per instruction), double-buffered
// on ASYNCcnt; accumulation is v_fmac from ds_load_b128; result stored with
// one global_store_b128 per lane. No f32 atomics anywhere.

__global__ void __launch_bounds__(AWPB * 32)
k_aggregate(const float* __restrict__ src,
            float* __restrict__ dst,
            const float* __restrict__ bias,
            const unsigned* __restrict__ offsets,
            const unsigned* __restrict__ src_idx,
            const float* __restrict__ wgt) {
  __shared__ float4 buf[AWPB][2][EB][32];   // 32 KB

  const int lane = threadIdx.x & 31;
  const int wv   = threadIdx.x >> 5;
  const int n    = blockIdx.x * AWPB + wv;
  if (n >= N_NODES) return;

  float4 acc = *(const float4*)(bias + lane * 4);   // out = segsum + bias

  const int beg = (int)offsets[n];
  const int end = (int)offsets[n + 1];
  const int cnt = end - beg;

  if (cnt > 0) {
    const int nb = (cnt + EB - 1) / EB;
    float wj_cur[EB], wj_nxt[EB];

    auto prefetch = [&](int chunk, int st, float* wj) {
      #pragma unroll
      for (int j = 0; j < EB; ++j) {
        int k = beg + chunk * EB + j;
        bool valid = (k < end);
        int kk = valid ? k : beg;           // clamp: uniform issue count per stage
        int r = (int)src_idx[kk];
        const float* gp = src + (size_t)r * HID + lane * 4;
        unsigned la = lds_addr_of(&buf[wv][st][j][lane]);
        asm volatile("global_load_async_to_lds_b128 %0, %1, off"
                     :: "v"(la), "v"(gp) : "memory");
        wj[j] = valid ? wgt[kk] : 0.0f;     // w=0 neutralizes clamped gathers
      }
    };
    auto consume = [&](int st, const float* wj) {
      #pragma unroll
      for (int j = 0; j < EB; ++j) {
        float4 v = buf[wv][st][j][lane];
        acc.x += wj[j] * v.x;  acc.y += wj[j] * v.y;
        acc.z += wj[j] * v.z;  acc.w += wj[j] * v.w;
      }
    };

    int stage = 0;
    prefetch(0, 0, wj_cur);
    for (int c = 0;;) {
      int cn = c + 1;
      bool more = (cn < nb);
      if (more) {
        prefetch(cn, stage ^ 1, wj_nxt);
        // async done-signals retire in order: <=4 left means old stage landed
        asm volatile("s_wait_asynccnt 4" ::: "memory");
      } else {
        asm volatile("s_wait_asynccnt 0" ::: "memory");
      }
      consume(stage, wj_cur);
      if (!more) break;
      #pragma unroll
      for (int j = 0; j < EB; ++j) wj_cur[j] = wj_nxt[j];
      c = cn;
      stage ^= 1;
    }
  }

  *(float4*)(dst + (size_t)n * HID + lane * 4) = acc;
}

// ---- CSR construction ----------------------------------------------------

__global__ void k_zero(unsigned* __restrict__ cnt) {
  int i = blockIdx.x * blockDim.x + threadIdx.x;
  if (i < N_NODES) cnt[i] = 0u;
}

__global__ void k_count(const int* __restrict__ eidx,
                        const int* __restrict__ pidx,
                        unsigned* __restrict__ cnt) {
  int i = blockIdx.x * blockDim.x + threadIdx.x;
  int stride = gridDim.x * blockDim.x;
  for (int e = i; e < E_TOTAL; e += stride) {
    int c = (e < NE) ? eidx[NE + e] : pidx[NP + (e - NE)];
    atomicAdd(&cnt[c], 1u);
  }
}

__global__ void k_dinv(const unsigned* __restrict__ cnt, float* __restrict__ dinv) {
  int i = blockIdx.x * blockDim.x + threadIdx.x;
  if (i < N_NODES) {
    unsigned c = cnt[i];
    dinv[i] = (c > 0u) ? __frsqrt_rn((float)c) : 0.0f;
  }
}

// Single-workgroup exclusive scan: cnt_cursor (counts) -> offsets[0..N] and
// cnt_cursor becomes the fill cursor (= exclusive offsets).
__global__ void __launch_bounds__(SCAN_T)
k_scan(unsigned* __restrict__ cnt_cursor, unsigned* __restrict__ offsets) {
  __shared__ unsigned s[SCAN_T];
  __shared__ unsigned carry_s;
  const int tid = threadIdx.x;
  if (tid == 0) carry_s = 0u;
  __syncthreads();

  for (int base = 0; base < N_NODES; base += SCAN_T) {
    int i = base + tid;
    unsigned v = (i < N_NODES) ? cnt_cursor[i] : 0u;
    s[tid] = v;
    __syncthreads();
    for (int off = 1; off < SCAN_T; off <<= 1) {   // Hillis-Steele inclusive
      unsigned t = (tid >= off) ? s[tid - off] : 0u;
      __syncthreads();
      s[tid] += t;
      __syncthreads();
    }
    unsigned excl = s[tid] - v + carry_s;
    if (i < N_NODES) { offsets[i] = excl; cnt_cursor[i] = excl; }
    unsigned total = s[SCAN_T - 1];
    __syncthreads();
    if (tid == 0) carry_s += total;
    __syncthreads();
  }
  if (tid == 0) offsets[N_NODES] = carry_s;   // = E_TOTAL
}

__global__ void k_fill(const int* __restrict__ eidx,
                       const int* __restrict__ pidx,
                       const float* __restrict__ dinv,
                       unsigned* __restrict__ cursor,
                       unsigned* __restrict__ src_idx,
                       float* __restrict__ wgt) {
  int i = blockIdx.x * blockDim.x + threadIdx.x;
  int stride = gridDim.x * blockDim.x;
  for (int e = i; e < E_TOTAL; e += stride) {
    int r, c;
    edge_rc(eidx, pidx, e, r, c);
    float w = dinv[r] * dinv[c];
    unsigned pos = atomicAdd(&cursor[c], 1u);
    src_idx[pos] = (unsigned)r;
    wgt[pos] = w;
  }
}

// ---- launch --------------------------------------------------------------

extern "C" void kernel_launch(void* const* d_in, const int* in_sizes, int n_in,
                              void* d_out, int out_size, void* d_ws, size_t ws_size,
                              hipStream_t stream) {
  const float* x     = (const float*)d_in[0];
  const float* bias0 = (const float*)d_in[1];
  const float* bias1 = (const float*)d_in[2];
  const int*   eidx  = (const int*)d_in[3];   // [2, NE] flat, int32
  const int*   pidx  = (const int*)d_in[4];   // [2, NP] flat, int32
  float* out = (float*)d_out;

  // workspace layout (4-byte units); h1 offset is 16B-aligned.
  float*    ws      = (float*)d_ws;
  float*    dinv    = ws;                                   // N
  unsigned* offsets = (unsigned*)(ws + N_NODES);            // N+1 (+pad to 4)
  unsigned* cursor  = (unsigned*)(ws + 2 * N_NODES + 4);    // N  (counts, then fill cursor)
  unsigned* src_idx = (unsigned*)(ws + 3 * N_NODES + 4);    // E_TOTAL
  float*    wgt     = ws + 3 * N_NODES + 4 + E_TOTAL;       // E_TOTAL
  float*    h1      = ws + 3 * N_NODES + 4 + 2 * E_TOTAL;   // N*HID (16B aligned)

  const int T = 256;
  const int nblk_nodes = (N_NODES + T - 1) / T;

  k_zero <<<nblk_nodes, T, 0, stream>>>(cursor);
  k_count<<<2048, T, 0, stream>>>(eidx, pidx, cursor);
  k_dinv <<<nblk_nodes, T, 0, stream>>>(cursor, dinv);
  k_scan <<<1, SCAN_T, 0, stream>>>(cursor, offsets);
  k_fill <<<2048, T, 0, stream>>>(eidx, pidx, dinv, cursor, src_idx, wgt);

  const int nblk_agg = (N_NODES + AWPB - 1) / AWPB;
  // layer 1: h1 = scatter-free pull of w*x[row] + bias0
  k_aggregate<<<nblk_agg, AWPB * 32, 0, stream>>>(x,  h1,  bias0, offsets, src_idx, wgt);
  // layer 2: out = pull of w*h1[row] + bias1
  k_aggregate<<<nblk_agg, AWPB * 32, 0, stream>>>(h1, out, bias1, offsets, src_idx, wgt);
}